// YOLOThreeOnLineValuationFunction_33646773797492
// MI455X (gfx1250) — compile-verified
//
#include <hip/hip_runtime.h>
#include <stdint.h>

// ---------------------------------------------------------------------------
// YOLO zone valuation: out[b] = dir[b, zone(z_1[b], z_2[b])]
// Memory-bound streaming kernel. Data movement uses the CDNA5 async
// global->LDS path (ASYNCcnt) with a double-buffered per-block pipeline.
// ---------------------------------------------------------------------------

#define TILE   256
#define ZDW    (TILE * 11)        // 2816 dwords per z chunk per tile
#define DDW    (TILE * 4)         // 1024 dwords per dir chunk per tile
#define BUFDW  (2 * ZDW + DDW)    // 6656 dwords per LDS buffer
#define Z1_B   0                  // byte offsets inside one LDS buffer
#define Z2_B   (ZDW * 4)
#define DIR_B  (2 * ZDW * 4)
#define BUF_B  (BUFDW * 4)        // 26624 bytes per buffer

__device__ __forceinline__ uint32_t umin32(uint32_t a, uint32_t b) {
  return a < b ? a : b;
}

// Async copy from global address sbase plus offset voff to LDS byte address lds_byte (GVS addressing, ASYNCcnt)
__device__ __forceinline__ void async_ld_b128(uint32_t lds_byte, uint64_t sbase,
                                              uint32_t voff) {
  asm volatile("global_load_async_to_lds_b128 %0, %1, %2"
               :: "v"(lds_byte), "v"(voff), "s"(sbase)
               : "memory");
}
__device__ __forceinline__ void async_ld_b32(uint32_t lds_byte, uint64_t sbase,
                                             uint32_t voff) {
  asm volatile("global_load_async_to_lds_b32 %0, %1, %2"
               :: "v"(lds_byte), "v"(voff), "s"(sbase)
               : "memory");
}

// Issue one tile's staging: 11 async ops per thread, uniform across waves.
__device__ __forceinline__ void issue_tile(uint64_t z1b, uint64_t z2b,
                                           uint64_t dirb, uint32_t tile,
                                           uint32_t bufbase, uint32_t t,
                                           uint32_t zc16, uint32_t zc4,
                                           uint32_t dc16) {
  const uint32_t zchunk = tile * (uint32_t)(ZDW * 4);  // byte offset in z arrays
  const uint32_t dchunk = tile * (uint32_t)(DDW * 4);  // byte offset in dir
  // Bulk: dwords [0, 2048) of the chunk as 16B transfers (2 per thread).
#pragma unroll
  for (uint32_t k = 0; k < 2; ++k) {
    const uint32_t o = (t + k * 256u) * 16u;  // 16B aligned
    async_ld_b128(bufbase + Z1_B + o, z1b, umin32(zchunk + o, zc16));
    async_ld_b128(bufbase + Z2_B + o, z2b, umin32(zchunk + o, zc16));
  }
  // Remainder: dwords [2048, 2816) as b32 transfers (3 per thread).
#pragma unroll
  for (uint32_t k = 0; k < 3; ++k) {
    const uint32_t o = (2048u + t + k * 256u) * 4u;
    async_ld_b32(bufbase + Z1_B + o, z1b, umin32(zchunk + o, zc4));
    async_ld_b32(bufbase + Z2_B + o, z2b, umin32(zchunk + o, zc4));
  }
  // dir chunk: 1024 dwords == one b128 per thread.
  {
    const uint32_t o = t * 16u;
    async_ld_b128(bufbase + DIR_B + o, dirb, umin32(dchunk + o, dc16));
  }
}

__global__ __launch_bounds__(TILE) void yolo_zone_kernel(
    const float* __restrict__ z1, const float* __restrict__ z2,
    const float* __restrict__ dirv, float* __restrict__ out, int n,
    int ntiles) {
  __shared__ float smem[2 * BUFDW];

  const uint32_t t = threadIdx.x;
  // Flat pointer to LDS: low 32 bits are the wave-relative LDS byte offset.
  const uint32_t smem_base = (uint32_t)(uintptr_t)(void*)&smem[0];
  const uint64_t z1b  = (uint64_t)(uintptr_t)z1;
  const uint64_t z2b  = (uint64_t)(uintptr_t)z2;
  const uint64_t dirb = (uint64_t)(uintptr_t)dirv;
  const uint32_t zbytes = (uint32_t)n * 44u;   // n*11*4
  const uint32_t dbytes = (uint32_t)n * 16u;   // n*4*4
  const uint32_t zc16 = zbytes - 16u;          // clamp for b128 (16B aligned)
  const uint32_t zc4  = zbytes - 4u;           // clamp for b32
  const uint32_t dc16 = dbytes - 16u;

  int tile = (int)blockIdx.x;
  if (tile >= ntiles) return;  // uniform per block

  uint32_t cur = 0;
  issue_tile(z1b, z2b, dirb, (uint32_t)tile, smem_base, t, zc16, zc4, dc16);

  for (;;) {
    const int next = tile + (int)gridDim.x;
    const bool have_next = next < ntiles;  // uniform per block
    if (have_next) {
      issue_tile(z1b, z2b, dirb, (uint32_t)next,
                 smem_base + (cur ^ 1u) * (uint32_t)BUF_B, t, zc16, zc4, dc16);
      // 11 newer ops outstanding; <=11 means current tile's 11 are done.
      asm volatile("s_wait_asynccnt 11" ::: "memory");
    } else {
      asm volatile("s_wait_asynccnt 0" ::: "memory");
    }
    __syncthreads();  // all waves' staging for `cur` complete

    const float* buf = &smem[cur * BUFDW];
    const float* bz1 = buf;
    const float* bz2 = buf + ZDW;
    const float* bd  = buf + 2 * ZDW;

    const float x0 = bz1[t * 11 + 0];
    const float y0 = bz1[t * 11 + 1];
    const float x1 = bz1[t * 11 + 2];
    const float y1 = bz1[t * 11 + 3];
    const float u0 = bz2[t * 11 + 0];
    const float v0 = bz2[t * 11 + 1];
    const float u1 = bz2[t * 11 + 2];
    const float v1 = bz2[t * 11 + 3];

    const float c1x = (x0 + x1) * 0.5f;
    const float c1y = (y0 + y1) * 0.5f;
    const float c2x = (u0 + u1) * 0.5f;
    const float c2y = (v0 + v1) * 0.5f;
    const float dx = c2x - c1x;
    const float dy = c1y - c2y;  // -(c2y - c1y)

    const float phi = atan2f(dy, dx) * 57.29577951308232f;  // degrees
    const int pl = (int)phi;                 // truncate toward zero
    int pcs = (90 - pl) % 360;               // python-style mod 360
    if (pcs < 0) pcs += 360;
    const int zone = ((pcs + 45) / 90) & 3;  // floor-div then mod 4

    const uint32_t r = (uint32_t)tile * TILE + t;
    if (r < (uint32_t)n) out[r] = bd[t * 4 + zone];

    __syncthreads();  // protect buffer `cur^1`... and `cur` before reuse
    if (!have_next) break;
    tile = next;
    cur ^= 1u;
  }
}

extern "C" void kernel_launch(void* const* d_in, const int* in_sizes, int n_in,
                              void* d_out, int out_size, void* d_ws,
                              size_t ws_size, hipStream_t stream) {
  (void)n_in; (void)d_ws; (void)ws_size; (void)out_size;
  const float* z1  = (const float*)d_in[0];
  const float* z2  = (const float*)d_in[1];
  // d_in[2] (z_3) is unused by the reference forward pass.
  const float* dir = (const float*)d_in[3];
  float* out = (float*)d_out;

  const int n = in_sizes[0] / 11;
  const int ntiles = (n + TILE - 1) / TILE;
  if (ntiles <= 0) return;
  int nb = ntiles < 2048 ? ntiles : 2048;

  yolo_zone_kernel<<<dim3((uint32_t)nb), dim3(TILE), 0, stream>>>(
      z1, z2, dir, out, n, ntiles);
}